// HeteroGraphSAGE_69415261438665
// MI455X (gfx1250) — compile-verified
//
#include <hip/hip_runtime.h>

// ---------------------------------------------------------------------------
// HeteroGraphSAGE (2 layers, weighted scatter-mean + residual + LayerNorm)
// MI455X strategy: memory/atomic-bound workload.
//  - scatter: coalesced 512B row gathers + hardware global_atomic_add_f32
//    (inline asm, scope DEV) into L2-resident accumulators (fits 192MB L2).
//  - finalize: LDS-staged rows; LayerNorm row statistics via
//    V_WMMA_F32_16X16X4_F32 (row-sum with B=ones, row-sumsq via Gram diag).
// ---------------------------------------------------------------------------

typedef __attribute__((ext_vector_type(2))) float v2f;
typedef __attribute__((ext_vector_type(8))) float v8f;

#define kNU 100000
#define kNI 100000
#define kE  1600000
#define kC  128

__device__ __forceinline__ void gadd_f32(float* p, float v) {
  // non-returning hardware f32 atomic add, device scope (L2 atomic units)
  asm volatile("global_atomic_add_f32 %0, %1, off scope:SCOPE_DEV"
               :: "v"(p), "v"(v) : "memory");
}

// One wave32 per edge; lane l covers channels [4l, 4l+4).
__global__ void __launch_bounds__(256)
sage_scatter(const float* __restrict__ xsrc, const int* __restrict__ ei,
             const float* __restrict__ ew, float* __restrict__ sums,
             float* __restrict__ cnt, int nedges) {
  const int wave = threadIdx.x >> 5;
  const int lane = threadIdx.x & 31;
  const int e = blockIdx.x * 8 + wave;
  if (e >= nedges) return;                       // wave-uniform
  const int s = ei[e];                           // src node
  const int d = ei[nedges + e];                  // dst node
  const float w = ew[e];
  const float4 x = ((const float4*)(xsrc + (size_t)s * kC))[lane];  // 512B coalesced
  float* p = sums + (size_t)d * kC + lane * 4;
  gadd_f32(p + 0, x.x * w);
  gadd_f32(p + 1, x.y * w);
  gadd_f32(p + 2, x.z * w);
  gadd_f32(p + 3, x.w * w);
  if (lane == 0) gadd_f32(cnt + d, 1.0f);
}

// One wave32 per 16 rows. Pass1: v = sums/max(cnt,1)+x -> LDS (coalesced).
// Pass2: WMMA f32 16x16x4: rowsum (B=ones) and rowsumsq (Gram diagonal;
//        A-fragment register layout == B^T layout per ISA 7.12.2).
// Pass3: normalize with gamma/beta (+optional relu), write out.
__global__ void __launch_bounds__(256)
sage_finalize_ln(const float* __restrict__ sums, const float* __restrict__ cnt,
                 const float* __restrict__ xdst, const float* __restrict__ g,
                 const float* __restrict__ b, float* __restrict__ out,
                 int n, int do_relu) {
  __shared__ float lds[8][16][kC + 4];           // +4 f32 row pad (bank spread)
  const int wave = threadIdx.x >> 5;
  const int lane = threadIdx.x & 31;
  const int tile = blockIdx.x * 8 + wave;
  if (tile * 16 >= n) return;                    // wave-uniform
  const int r0 = tile * 16;
  float (*T)[kC + 4] = lds[wave];

  // ---- Pass 1: build v rows in LDS, fully coalesced global loads ----
#pragma unroll
  for (int m = 0; m < 16; ++m) {
    int row = r0 + m; if (row > n - 1) row = n - 1;       // clamp (keep EXEC full)
    const float4 s4 = ((const float4*)(sums + (size_t)row * kC))[lane];
    const float4 x4 = ((const float4*)(xdst + (size_t)row * kC))[lane];
    float c = cnt[row]; c = fmaxf(c, 1.0f);
    const float inv = __frcp_rn(c);
    float4 v;
    v.x = s4.x * inv + x4.x;  v.y = s4.y * inv + x4.y;
    v.z = s4.z * inv + x4.z;  v.w = s4.w * inv + x4.w;
    *(float4*)&T[m][4 * lane] = v;
  }
  __syncthreads();

  // ---- Pass 2: WMMA statistics (EXEC all-ones, wave-uniform control) ----
  v8f rsum = {};      // D = A x ones : every column = row sum
  v8f gram = {};      // D = A x A^T : diagonal = row sum of squares
  v2f ones; ones.x = 1.0f; ones.y = 1.0f;
  const int mrow  = lane & 15;
  const int khalf = (lane >> 4) * 2;             // K = {0,1} lanes 0-15, {2,3} lanes 16-31
  for (int k0 = 0; k0 < kC; k0 += 4) {
    const v2f a = *(const v2f*)&T[mrow][k0 + khalf];   // ds_load_b64, conflict-free
    rsum = __builtin_amdgcn_wmma_f32_16x16x4_f32(false, a, false, ones,
                                                 (short)0, rsum, false, false);
    gram = __builtin_amdgcn_wmma_f32_16x16x4_f32(false, a, false, a,
                                                 (short)0, gram, false, false);
  }

  // ---- Pass 3: normalize + affine (+relu), coalesced stores ----
  const float4 g4 = ((const float4*)g)[lane];
  const float4 b4 = ((const float4*)b)[lane];
#pragma unroll
  for (int m = 0; m < 16; ++m) {
    // C/D layout: vgpr r, lanes0-15: M=r,N=lane ; lanes16-31: M=r+8,N=lane-16
    const float rs = __shfl(rsum[m & 7], (m < 8) ? 0 : 16, 32);
    const float sq = __shfl(gram[m & 7], (m < 8) ? m : (m + 16), 32);
    const float mean = rs * (1.0f / kC);
    const float var  = sq * (1.0f / kC) - mean * mean;
    const float rstd = rsqrtf(var + 1e-5f);
    const float4 v = *(const float4*)&T[m][4 * lane];
    float4 o;
    o.x = (v.x - mean) * rstd * g4.x + b4.x;
    o.y = (v.y - mean) * rstd * g4.y + b4.y;
    o.z = (v.z - mean) * rstd * g4.z + b4.z;
    o.w = (v.w - mean) * rstd * g4.w + b4.w;
    if (do_relu) {
      o.x = fmaxf(o.x, 0.0f); o.y = fmaxf(o.y, 0.0f);
      o.z = fmaxf(o.z, 0.0f); o.w = fmaxf(o.w, 0.0f);
    }
    const int row = r0 + m;
    if (row < n)                                  // wave-uniform guard
      ((float4*)(out + (size_t)row * kC))[lane] = o;
  }
}

extern "C" void kernel_launch(void* const* d_in, const int* in_sizes, int n_in,
                              void* d_out, int out_size, void* d_ws, size_t ws_size,
                              hipStream_t stream) {
  const float* x_user = (const float*)d_in[0];
  const float* x_item = (const float*)d_in[1];
  const float* ew_ui  = (const float*)d_in[2];
  const float* ew_iu  = (const float*)d_in[3];
  const float* lnwu0  = (const float*)d_in[4];
  const float* lnbu0  = (const float*)d_in[5];
  const float* lnwu1  = (const float*)d_in[6];
  const float* lnbu1  = (const float*)d_in[7];
  const float* lnwi0  = (const float*)d_in[8];
  const float* lnbi0  = (const float*)d_in[9];
  const float* lnwi1  = (const float*)d_in[10];
  const float* lnbi1  = (const float*)d_in[11];
  const int*   ei_ui  = (const int*)d_in[12];
  const int*   ei_iu  = (const int*)d_in[13];
  float* out = (float*)d_out;

  // workspace layout: sums_u | sums_i | cnt_u | cnt_i  (~103 MB, L2-resident hot set)
  float* sums_u = (float*)d_ws;
  float* sums_i = sums_u + (size_t)kNU * kC;
  float* cnt_u  = sums_i + (size_t)kNI * kC;
  float* cnt_i  = cnt_u + kNU;
  const size_t zbytes =
      ((size_t)kNU * kC + (size_t)kNI * kC + kNU + kNI) * sizeof(float);

  float* outU = out;                    // layer-0 user result / layer-1 input
  float* outI = out + (size_t)kNU * kC; // layer-0 item result / layer-1 input

  const dim3 blk(256);
  const int sgrid = (kE + 7) / 8;
  const int fgU = ((kNU + 15) / 16 + 7) / 8;
  const int fgI = ((kNI + 15) / 16 + 7) / 8;

  // ----- layer 0 -----
  hipMemsetAsync(d_ws, 0, zbytes, stream);
  sage_scatter<<<sgrid, blk, 0, stream>>>(x_item, ei_iu, ew_iu, sums_u, cnt_u, kE);
  sage_scatter<<<sgrid, blk, 0, stream>>>(x_user, ei_ui, ew_ui, sums_i, cnt_i, kE);
  sage_finalize_ln<<<fgU, blk, 0, stream>>>(sums_u, cnt_u, x_user, lnwu0, lnbu0,
                                            outU, kNU, 1);
  sage_finalize_ln<<<fgI, blk, 0, stream>>>(sums_i, cnt_i, x_item, lnwi0, lnbi0,
                                            outI, kNI, 1);
  // ----- layer 1 (in place on d_out halves) -----
  hipMemsetAsync(d_ws, 0, zbytes, stream);
  sage_scatter<<<sgrid, blk, 0, stream>>>(outI, ei_iu, ew_iu, sums_u, cnt_u, kE);
  sage_scatter<<<sgrid, blk, 0, stream>>>(outU, ei_ui, ew_ui, sums_i, cnt_i, kE);
  sage_finalize_ln<<<fgU, blk, 0, stream>>>(sums_u, cnt_u, outU, lnwu1, lnbu1,
                                            outU, kNU, 0);
  sage_finalize_ln<<<fgI, blk, 0, stream>>>(sums_i, cnt_i, outI, lnwi1, lnbi1,
                                            outI, kNI, 0);
}